// GCNLayer_27444841022058
// MI455X (gfx1250) — compile-verified
//
#include <hip/hip_runtime.h>

typedef float v2f __attribute__((ext_vector_type(2)));
typedef float v8f __attribute__((ext_vector_type(8)));

#define D_DIM 128            // D_IN == D_OUT == 128
#define ROWS_PER_BLOCK 64    // rows of the node matrix per workgroup
#define GEMM_THREADS 128     // 4 waves (wave32)
#define LDS_STRIDE 130       // pad 128 -> 130 floats: avoids 16-way LDS bank conflicts

// ---------------------------------------------------------------------------
// Kernel 1: zero the accumulator (d_out is poisoned by the harness).
// ---------------------------------------------------------------------------
__global__ void zero_out_kernel(float* __restrict__ out, size_t n) {
    size_t i = (size_t)blockIdx.x * blockDim.x + threadIdx.x;
    size_t stride = (size_t)gridDim.x * blockDim.x;
    for (; i < n; i += stride) out[i] = 0.0f;
}

// ---------------------------------------------------------------------------
// Kernel 2: edge scatter  out[dst] += w * x[src]
// One wave32 per edge; each lane owns a contiguous float4 slice of the
// 128-wide feature row. x (51.2 MB) and out are L2-resident (192 MB L2),
// so both the gather and the atomics hit L2.
// unsafeAtomicAdd -> global_atomic_add_f32 (hardware FP atomic, no CAS loop).
// ---------------------------------------------------------------------------
__global__ __launch_bounds__(256)
void gcn_scatter_kernel(const float* __restrict__ x,
                        const int*   __restrict__ esrc,
                        const int*   __restrict__ edst,
                        const float* __restrict__ ew,
                        float* out, int n_edges) {
    int gid  = blockIdx.x * blockDim.x + threadIdx.x;
    int e    = gid >> 5;        // one wave32 per edge
    int lane = gid & 31;        // lane -> 4 floats of the 128-wide row
    if (e >= n_edges) return;

    int   s  = esrc[e];
    int   d  = edst[e];
    float wt = ew[e];

    const float4 v = *(const float4*)(x + (size_t)s * D_DIM + lane * 4);
    float* o = out + (size_t)d * D_DIM + lane * 4;
    unsafeAtomicAdd(o + 0, v.x * wt);
    unsafeAtomicAdd(o + 1, v.y * wt);
    unsafeAtomicAdd(o + 2, v.z * wt);
    unsafeAtomicAdd(o + 3, v.w * wt);
}

// ---------------------------------------------------------------------------
// Kernel 3: in-place GEMM  out = out @ W.T  using V_WMMA_F32_16X16X4_F32.
// Block = 128 threads = 4 waves. Block stages its 64 rows into LDS (so the
// in-place overwrite is safe: all reads of those rows happen before any
// store), then each wave computes a 16x128 slab as 8 WMMA tiles with a
// K-loop of 32 chained 16x16x4 f32 WMMAs (f32 accumulate, full precision).
//
// Fragment layouts (ISA 7.12.2, wave32):
//   A 16x4 f32 : lane = 16*khalf + m ; VGPR0/1 = K = 2*khalf + {0,1}
//   B 4x16 f32 : lane = 16*khalf + n ; VGPR0/1 = K = 2*khalf + {0,1}
//                (B[k][n] = W[n*D + k] since h = x @ W.T)
//   C/D 16x16  : VGPR r holds M = r + 8*(lane/16), N = lane%16
// ---------------------------------------------------------------------------
__global__ __launch_bounds__(GEMM_THREADS)
void gcn_gemm_wmma_kernel(float* inout,               // aggregated rows, overwritten
                          const float* __restrict__ W,
                          int n_nodes) {
    __shared__ float xs[ROWS_PER_BLOCK * LDS_STRIDE];

    const int tid        = threadIdx.x;
    const int wave       = tid >> 5;
    const int lane       = tid & 31;
    const int block_row0 = blockIdx.x * ROWS_PER_BLOCK;

    // ---- Stage this block's 64x128 row tile into LDS (coalesced float4) ----
    for (int i = tid; i < ROWS_PER_BLOCK * (D_DIM / 4); i += GEMM_THREADS) {
        int r    = i / (D_DIM / 4);
        int c4   = i % (D_DIM / 4);
        int grow = block_row0 + r;
        float4 v = make_float4(0.f, 0.f, 0.f, 0.f);
        if (grow < n_nodes)
            v = *(const float4*)(inout + (size_t)grow * D_DIM + c4 * 4);
        float* dst = &xs[r * LDS_STRIDE + c4 * 4];
        dst[0] = v.x; dst[1] = v.y; dst[2] = v.z; dst[3] = v.w;
    }
    __syncthreads();

    // ---- Each wave: rows [wave*16, wave*16+16) x all 128 output columns ----
    const int mrow    = wave * 16;     // local row base of this wave's slab
    const int lane_m  = lane & 15;     // A: row-in-tile, B: col-in-tile, C: N
    const int lane_hi = lane >> 4;     // A/B: K-pair select, C: M-half select

    v8f acc[8];
    #pragma unroll
    for (int t = 0; t < 8; ++t) acc[t] = (v8f){};

    for (int k = 0; k < D_DIM; k += 4) {
        const int ka = k + 2 * lane_hi;
        // A fragment from LDS (stride-130 padding -> conflict-free ds_load_b64)
        v2f a = *(const v2f*)&xs[(mrow + lane_m) * LDS_STRIDE + ka];
        #pragma unroll
        for (int t = 0; t < 8; ++t) {
            // B fragment: B[k][o] = W[o*D + k]; W is 64 KB -> L2/WGP$-hot
            v2f b = *(const v2f*)(W + (size_t)(t * 16 + lane_m) * D_DIM + ka);
            acc[t] = __builtin_amdgcn_wmma_f32_16x16x4_f32(
                /*neg_a=*/false, a, /*neg_b=*/false, b,
                /*c_mod=*/(short)0, acc[t], /*reuse_a=*/false, /*reuse_b=*/false);
        }
    }
    __syncthreads();   // all LDS reads done before anyone could ever re-stage

    // ---- Store the 16x128 slab back in place ----
    const int grow_base = block_row0 + mrow;
    #pragma unroll
    for (int t = 0; t < 8; ++t) {
        #pragma unroll
        for (int vr = 0; vr < 8; ++vr) {
            int grow = grow_base + vr + 8 * lane_hi;
            if (grow < n_nodes)
                inout[(size_t)grow * D_DIM + t * 16 + lane_m] = acc[t][vr];
        }
    }
}

// ---------------------------------------------------------------------------
// Launch: since the layer is linear, segment_sum(w * (xW^T)[src]) ==
// (segment_sum(w * x[src])) @ W^T.  Aggregate first (into d_out), then run
// the WMMA GEMM in place on d_out.  No workspace needed.
// ---------------------------------------------------------------------------
extern "C" void kernel_launch(void* const* d_in, const int* in_sizes, int n_in,
                              void* d_out, int out_size, void* d_ws, size_t ws_size,
                              hipStream_t stream) {
    const float* x    = (const float*)d_in[0];
    const float* W    = (const float*)d_in[1];
    const int*   esrc = (const int*)d_in[2];
    const int*   edst = (const int*)d_in[3];
    const float* ew   = (const float*)d_in[4];
    float*       out  = (float*)d_out;

    const int n_nodes = in_sizes[0] / D_DIM;
    const int n_edges = in_sizes[2];

    // 1) zero accumulator
    zero_out_kernel<<<1024, 256, 0, stream>>>(out, (size_t)out_size);

    // 2) edge scatter: one wave per edge
    long long sthreads = (long long)n_edges * 32;
    int sblocks = (int)((sthreads + 255) / 256);
    gcn_scatter_kernel<<<sblocks, 256, 0, stream>>>(x, esrc, edst, ew, out, n_edges);

    // 3) in-place WMMA GEMM: out = out @ W.T
    int gblocks = (n_nodes + ROWS_PER_BLOCK - 1) / ROWS_PER_BLOCK;
    gcn_gemm_wmma_kernel<<<gblocks, GEMM_THREADS, 0, stream>>>(out, W, n_nodes);
}